// Compute_NH_Potentials_9543417332555
// MI455X (gfx1250) — compile-verified
//
#include <hip/hip_runtime.h>
#include <hip/hip_bf16.h>
#include <stdint.h>

// Neo-Hookean constants: E = 1.0, nu = 0.3
// MU  = E / (2 (1+nu))            = 1/2.6
// LAM = E nu / ((1+nu)(1-2nu))    = 0.3/0.52
#define MU_NH   0.38461538461538464f
#define LAM_NH  0.57692307692307687f

#define THREADS        256
#define TILE_M         1024                   // matrices per block
#define MATS_PER_THR   (TILE_M / THREADS)     // 4
#define TILE_FLOATS    (TILE_M * 9)           // 9216 floats = 36 KB
#define TILE_F4        (TILE_FLOATS / 4)      // 2304 float4 chunks
#define F4_PER_THR     (TILE_F4 / THREADS)    // 9 async B128 copies per thread

// CDNA5 async global->LDS copy. IOFFSET is applied to BOTH the global address
// and the LDS address,
// so one VADDR pair + one LDS-offset VGPR serve all 9 copies;
// only the immediate varies (stride 4096 B = 256 lanes * 16 B).
#define ASYNC_CP_OFF(OFFBYTES)                                                 \
  asm volatile("global_load_async_to_lds_b128 %0, %1, off offset:" #OFFBYTES   \
               :                                                               \
               : "v"(lds0), "v"(g0)                                            \
               : "memory")

// Guarded single copy for the (single) partial tail tile.
__device__ __forceinline__ void async_copy_b128(uint32_t lds_byte_off,
                                                const void* gaddr) {
  asm volatile("global_load_async_to_lds_b128 %0, %1, off"
               :
               : "v"(lds_byte_off), "v"(gaddr)
               : "memory");
}

__device__ __forceinline__ void wait_async_all() {
  asm volatile("s_wait_asynccnt 0" ::: "memory");
}

__device__ __forceinline__ void nh_compute_store(const float* __restrict__ smem,
                                                 float* __restrict__ w,
                                                 int ml, int m) {
  const float* a = &smem[ml * 9];
  const float a00 = a[0], a01 = a[1], a02 = a[2];
  const float a10 = a[3], a11 = a[4], a12 = a[5];
  const float a20 = a[6], a21 = a[7], a22 = a[8];

  // I1 = trace(X^T X) = Frobenius norm squared
  const float I1 = a00*a00 + a01*a01 + a02*a02
                 + a10*a10 + a11*a11 + a12*a12
                 + a20*a20 + a21*a21 + a22*a22;

  // J = det(X)
  const float J = a00 * (a11*a22 - a12*a21)
                - a01 * (a10*a22 - a12*a20)
                + a02 * (a10*a21 - a11*a20);

  const float lj = __logf(J);

  const float W = 0.5f * MU_NH * (I1 - 3.0f)
                - MU_NH * lj
                + 0.5f * LAM_NH * lj * lj;

  __builtin_nontemporal_store(W, &w[m]);   // write-once stream
}

__global__ __launch_bounds__(THREADS)
void nh_potential_kernel(const float* __restrict__ x,
                         float* __restrict__ w,
                         int n) {
  __shared__ __align__(16) float smem[TILE_FLOATS];

  const int tid   = threadIdx.x;
  const int tile0 = (int)blockIdx.x * TILE_M;          // first matrix of tile
  const bool full = (tile0 + TILE_M) <= n;             // uniform per block

  // Generic pointer to LDS carries the LDS byte offset in its low 32 bits
  // (ISA 10.2 aperture mapping).
  const uint32_t lds0   = (uint32_t)(uintptr_t)&smem[0] + (uint32_t)tid * 16u;
  const void*    g0     = (const void*)(x + (size_t)tile0 * 9 + tid * 4);

  if (full) {
    // ---- Fast path: 9 back-to-back async B128 copies, no masking ----
    ASYNC_CP_OFF(0);
    ASYNC_CP_OFF(4096);
    ASYNC_CP_OFF(8192);
    ASYNC_CP_OFF(12288);
    ASYNC_CP_OFF(16384);
    ASYNC_CP_OFF(20480);
    ASYNC_CP_OFF(24576);
    ASYNC_CP_OFF(28672);
    ASYNC_CP_OFF(32768);
  } else {
    // ---- Tail tile (one block): per-copy float4 bounds guard ----
    const int total_f = n * 9;                          // fits in int32
    const int base_f  = tile0 * 9;
#pragma unroll
    for (int k = 0; k < F4_PER_THR; ++k) {
      const int j  = tid + k * THREADS;                 // float4 index in tile
      const int gf = base_f + j * 4;                    // global float index
      if (gf + 3 < total_f) {
        async_copy_b128((uint32_t)(uintptr_t)&smem[0] + (uint32_t)j * 16u,
                        (const void*)(x + gf));
      }
    }
  }

  wait_async_all();     // this wave's async transfers landed in LDS
  __syncthreads();      // make every wave's staged data visible to the block

  // ---- Compute: stride-9 LDS gather is bank-conflict free (gcd(9,64)=1) ----
  if (full) {
#pragma unroll
    for (int q = 0; q < MATS_PER_THR; ++q) {
      const int ml = q * THREADS + tid;
      nh_compute_store(smem, w, ml, tile0 + ml);
    }
  } else {
#pragma unroll
    for (int q = 0; q < MATS_PER_THR; ++q) {
      const int ml = q * THREADS + tid;
      const int m  = tile0 + ml;
      if (m < n) nh_compute_store(smem, w, ml, m);
    }
  }
}

extern "C" void kernel_launch(void* const* d_in, const int* in_sizes, int n_in,
                              void* d_out, int out_size, void* d_ws, size_t ws_size,
                              hipStream_t stream) {
  const float* x = (const float*)d_in[0];
  float* w = (float*)d_out;
  const int n = in_sizes[0] / 9;                       // number of 3x3 matrices
  const int blocks = (n + TILE_M - 1) / TILE_M;        // 3907 for B = 4e6
  hipLaunchKernelGGL(nh_potential_kernel, dim3(blocks), dim3(THREADS), 0, stream,
                     x, w, n);
}